// YOLODetector_15006615732558
// MI455X (gfx1250) — compile-verified
//
#include <hip/hip_runtime.h>

// ---- problem constants (from reference) ----
#define BB     32
#define AA     3
#define NCC    80
#define NOO    85          // NC + 5
#define GG     52
#define TT     512
#define STRIDE 8.0f        // 416 / 52
#define GG2    (GG*GG)                 // 2704
#define NCELL  (BB*AA*GG2)             // 259584
#define TC     128                     // cells per tile (LDS stage)
#define NTILE  ((GG2 + TC - 1)/TC)     // 22
#define PRED_N ((size_t)NCELL*NOO)     // 22,064,640

#if defined(__has_builtin)
#if __has_builtin(__builtin_amdgcn_tensor_load_to_lds) && __has_builtin(__builtin_amdgcn_s_wait_tensorcnt)
#define USE_TDM 1
#endif
#endif
#ifndef USE_TDM
#define USE_TDM 0
#endif

#if USE_TDM
typedef unsigned int u32x4 __attribute__((ext_vector_type(4)));
typedef int          i32x4 __attribute__((ext_vector_type(4)));
typedef int          i32x8 __attribute__((ext_vector_type(8)));
#endif

__device__ __forceinline__ float sigmoidf_(float z) {
    return 1.0f / (1.0f + __expf(-z));
}
// log(1 + e^z), numerically stable
__device__ __forceinline__ float softplusf_(float z) {
    return fmaxf(z, 0.0f) + log1pf(__expf(-fabsf(z)));
}

// ---------------- init: zero workspace ----------------
__global__ void k_init(float* ws, int n) {
    int i = blockIdx.x * blockDim.x + threadIdx.x;
    int stride = gridDim.x * blockDim.x;
    for (; i < n; i += stride) ws[i] = 0.0f;
}

// ---------------- targets: scatter masks / truth ----------------
__global__ void k_targets(const float* __restrict__ tg,
                          const float* __restrict__ anchors,
                          int* __restrict__ objm, int* __restrict__ ignm,
                          int* __restrict__ tclsA, float* __restrict__ tbox) {
    int t = threadIdx.x;
    if (t >= TT) return;
    const float* r = tg + t * 6;
    int   img  = (int)r[0];
    int   tcls = (int)r[1];
    float x1 = r[2] / STRIDE, y1 = r[3] / STRIDE;
    float x2 = r[4] / STRIDE, y2 = r[5] / STRIDE;
    float bw = x2 - x1, bh = y2 - y1;
    float cx = x1 + bw * 0.5f, cy = y1 + bh * 0.5f;

    float iou[AA];
    float best = -1.0f; int ba = 0;
#pragma unroll
    for (int a = 0; a < AA; ++a) {
        float aw = anchors[2*a] / STRIDE, ah = anchors[2*a+1] / STRIDE;
        float inter = fminf(aw, bw) * fminf(ah, bh);
        float uni   = aw*ah + bw*bh - inter;
        iou[a] = inter / uni;
        if (iou[a] > best) { best = iou[a]; ba = a; }   // argmax, first-max wins
    }
    int gi = (int)floorf(cx), gj = (int)floorf(cy);
    gi = min(max(gi, 0), GG-1); gj = min(max(gj, 0), GG-1);
    float tx = cx - floorf(cx), ty = cy - floorf(cy);   // before clamp, as in ref
    float awb = anchors[2*ba] / STRIDE, ahb = anchors[2*ba+1] / STRIDE;
    float tw = logf(bw / awb + 1e-10f);
    float th = logf(bh / ahb + 1e-10f);

    int ci = ((img*AA + ba)*GG + gj)*GG + gi;
    objm[ci]  = 1;
    tclsA[ci] = tcls;
    tbox[ci*4+0] = tx; tbox[ci*4+1] = ty;
    tbox[ci*4+2] = tw; tbox[ci*4+3] = th;
#pragma unroll
    for (int a = 0; a < AA; ++a)
        if (iou[a] > 0.5f)
            ignm[((img*AA + a)*GG + gj)*GG + gi] = 1;
}

// ---------------- main: TDM-stage tile, preds + loss partials ----------------
__global__ void __launch_bounds__(TC)
k_main(const float* __restrict__ x, const float* __restrict__ anchors,
       const int* __restrict__ objm, const int* __restrict__ ignm,
       const int* __restrict__ tclsA, const float* __restrict__ tbox,
       float* __restrict__ out, float* __restrict__ acc) {
    __shared__ __align__(16) float tile[NOO * TC];   // 43,520 B

    const int blk     = blockIdx.x;
    const int tileIdx = blk % NTILE;
    const int baIdx   = blk / NTILE;           // 0..95
    const int a       = baIdx % AA;
    const int b       = baIdx / AA;
    const int cell0   = tileIdx * TC;
    const int t       = threadIdx.x;
    const int lane    = t & 31;
    const int wave    = t >> 5;                // 4 waves of 32 (wave32)

    const unsigned lds_base = (unsigned)(unsigned long long)(&tile[0]);
    const float*   xb       = x + ((size_t)(b * (AA * NOO) + a * NOO)) * GG2 + cell0;

#if USE_TDM
    // ---- Tensor Data Mover: one descriptor loads the whole 85x128 tile ----
    // D# per CDNA5 ISA:
    //   2-D tile: tile_dim0 = TC cells (contiguous), tile_dim1 = NOO channel rows,
    //   row stride = GG2 elements, data_size = 4B, rows packed in LDS (pad off).
    //   tensor_dim0 = cells remaining -> OOB columns of the ragged last tile read 0.
    if (wave == 0) {
        unsigned long long ga = (unsigned long long)(const void*)xb;
        const int td0 = GG2 - cell0;                   // remaining cells in grid
        u32x4 g0;
        g0.x = 0x1u;                                   // count=1 (valid descriptor)
        g0.y = lds_base;                               // lds_addr (bytes)
        g0.z = (unsigned)(ga & 0xFFFFFFFFull);         // global_addr[31:0]
        g0.w = (unsigned)((ga >> 32) & 0x01FFFFFFull)  // global_addr[56:32]
             | 0x80000000u;                            // type=2 ("image")
        i32x8 g1;
        g1[0] = 0x00020000;                            // data_size=2 (4 bytes)
        g1[1] = (td0 & 0xFFFF) << 16;                  // tensor_dim0[15:0] @ [63:48]
        g1[2] = (int)(((unsigned)td0 >> 16)            // tensor_dim0[31:16]
              | ((unsigned)NOO << 16));                // tensor_dim1=85 @ [95:80]
        g1[3] = (TC << 16);                            // tile_dim0=128 @ [127:112]
        g1[4] = NOO;                                   // tile_dim1=85 @ [143:128]
        g1[5] = GG2;                                   // tensor_dim0_stride=2704
        g1[6] = 0;
        g1[7] = 0;
        i32x4 z4 = {0, 0, 0, 0};                       // groups 2/3 unused (2-D)
        i32x8 z8 = {0, 0, 0, 0, 0, 0, 0, 0};           // extra group (clang-23 6-arg form)
        __builtin_amdgcn_tensor_load_to_lds(g0, g1, z4, z4, z8, 0);
    }
    __builtin_amdgcn_s_wait_tensorcnt(0);
    __syncthreads();
#else
    // ---- fallback: per-wave async global->LDS staging ----
    const unsigned voff = (unsigned)lane * 16u;
    const bool     inb  = (cell0 + lane * 4) < GG2;
    for (int c = 0; c < NOO; ++c) {            // c is block-uniform -> saddr stays SGPR
        if (((c & 3) == wave) && inb) {
            const float* sbase = xb + (size_t)c * GG2;
            unsigned laddr = lds_base + (unsigned)c * (TC * 4) + voff;
            asm volatile("global_load_async_to_lds_b128 %0, %1, %2"
                         :: "v"(laddr), "v"(voff), "s"(sbase)
                         : "memory");
        }
    }
    asm volatile("s_wait_asynccnt 0" ::: "memory");
    __syncthreads();
#endif

    // ---- per-cell compute ----
    const int nc = min(TC, GG2 - cell0);
    float vlx=0.f,vly=0.f,vlw=0.f,vlh=0.f,vso=0.f,vsn=0.f,vsc=0.f,vno=0.f,vnn=0.f;

    if (t < nc) {
        const int cell = cell0 + t;
        const int g1i = cell / GG, g2i = cell % GG;
        const int ci = (b * AA + a) * GG2 + cell;
        const float aw = anchors[2*a], ah = anchors[2*a+1];

        const float ch0 = tile[0*TC + t], ch1 = tile[1*TC + t];
        const float ch2 = tile[2*TC + t], ch3 = tile[3*TC + t];
        const float ch4 = tile[4*TC + t];
        const float s0 = sigmoidf_(ch0), s1 = sigmoidf_(ch1), s4 = sigmoidf_(ch4);

        float* o = out + (size_t)ci * NOO;
        o[0] = (s0 + (float)g1i) * STRIDE;
        o[1] = (s1 + (float)g2i) * STRIDE;
        o[2] = __expf(ch2) * aw;               // exp(wh) * anchors_g * stride == * anchors
        o[3] = __expf(ch3) * ah;
        o[4] = s4;
#pragma unroll 4
        for (int c = 0; c < NCC; ++c)
            o[5 + c] = sigmoidf_(tile[(5 + c)*TC + t]);

        const bool isobj = objm[ci] != 0;
        const bool ign   = ignm[ci] != 0;
        if (isobj) {
            const float* tb = tbox + (size_t)ci * 4;
            float dx = s0  - tb[0], dy = s1  - tb[1];
            float dw = ch2 - tb[2], dh = ch3 - tb[3];
            vlx = dx*dx; vly = dy*dy; vlw = dw*dw; vlh = dh*dh;
            vso = fminf(softplusf_(-ch4), 100.0f);   // -clip(log(sigmoid),-100)
            vno = 1.0f;
            const int tc_ = tclsA[ci];
            float sc = 0.0f;
            for (int c = 0; c < NCC; ++c) {
                float z = tile[(5 + c)*TC + t];
                sc += (c == tc_) ? fminf(softplusf_(-z), 100.0f)
                                 : fminf(softplusf_( z), 100.0f);
            }
            vsc = sc;
        } else if (!ign) {
            vsn = fminf(softplusf_(ch4), 100.0f);    // -clip(log(1-sigmoid),-100)
            vnn = 1.0f;
        }
    }

    // ---- wave32 reduction + global accumulate ----
    float vals[9] = {vlx, vly, vlw, vlh, vso, vsn, vsc, vno, vnn};
#pragma unroll
    for (int i = 0; i < 9; ++i) {
        float v = vals[i];
#pragma unroll
        for (int off = 16; off > 0; off >>= 1)
            v += __shfl_xor(v, off, 32);
        if (lane == 0 && v != 0.0f) atomicAdd(&acc[i], v);
    }
}

// ---------------- finalize: scalar loss ----------------
__global__ void k_final(const float* __restrict__ acc, float* __restrict__ out) {
    if (threadIdx.x == 0 && blockIdx.x == 0) {
        float nobj = fmaxf(acc[7], 1.0f);
        float nno  = fmaxf(acc[8], 1.0f);
        float loss = (acc[0] + acc[1] + acc[2] + acc[3] + acc[4]) / nobj
                   + 100.0f * acc[5] / nno
                   + acc[6] / (nobj * (float)NCC);
        out[PRED_N] = loss;
    }
}

extern "C" void kernel_launch(void* const* d_in, const int* in_sizes, int n_in,
                              void* d_out, int out_size, void* d_ws, size_t ws_size,
                              hipStream_t stream) {
    const float* x       = (const float*)d_in[0];
    const float* targets = (const float*)d_in[1];
    const float* anchors = (const float*)d_in[2];
    float*       out     = (float*)d_out;
    float*       ws      = (float*)d_ws;

    int*   objm  = (int*)ws;
    int*   ignm  = objm  + NCELL;
    int*   tclsA = ignm  + NCELL;
    float* tbox  = (float*)(tclsA + NCELL);
    float* acc   = tbox + (size_t)NCELL * 4;
    const int nwords = NCELL * 7 + 16;

    k_init   <<<512, 256, 0, stream>>>(ws, nwords);
    k_targets<<<1,   TT,  0, stream>>>(targets, anchors, objm, ignm, tclsA, tbox);
    k_main   <<<BB * AA * NTILE, TC, 0, stream>>>(x, anchors, objm, ignm, tclsA,
                                                  tbox, out, acc);
    k_final  <<<1, 1, 0, stream>>>(acc, out);
}